// IdentityFMap_7937099563509
// MI455X (gfx1250) — compile-verified
//
#include <hip/hip_runtime.h>

typedef __attribute__((ext_vector_type(2))) float v2f;
typedef __attribute__((ext_vector_type(8))) float v8f;

#define B_ 2
#define N_ 7000
#define K_ 128
#define NT_ 7008          // padded N (multiple of 32)
#define YT32_ 219         // NT/32 y-strips for argmin
#define XT_ 438           // ceil(N/16) x-tiles
#define SLICES_ 8         // x-range slices for argmin parallelism
#define TL_ 219           // NT/32 tiles for big GEMM
#define BIGPEN_ 1.0e10f

__device__ __forceinline__ v8f wmma4(v2f a, v2f b, v8f c) {
    // D = A(16x4 f32) * B(4x16 f32) + C(16x16 f32)
    return __builtin_amdgcn_wmma_f32_16x16x4_f32(
        false, a, false, b, (short)0, c, false, false);
}

__device__ __forceinline__ unsigned int fkey(float f) {
    unsigned int u = __float_as_uint(f);
    return u ^ ((u >> 31) ? 0xFFFFFFFFu : 0x80000000u);  // order-preserving
}

// sx[b,x] = ||ex[b,x,:]||^2 + (1-mask_x)*BIG ; init packed argmin buffer
__global__ void k_prep(const float* __restrict__ ex, const float* __restrict__ mx,
                       float* __restrict__ sx, unsigned long long* __restrict__ packed) {
    int i = blockIdx.x * blockDim.x + threadIdx.x;
    if (i >= B_ * N_) return;
    const float* row = ex + (size_t)i * K_;
    float s = 0.0f;
    for (int k = 0; k < K_; k += 4) {
        float4 v = *(const float4*)(row + k);
        s += v.x * v.x + v.y * v.y + v.z * v.z + v.w * v.w;
    }
    sx[i] = s + (1.0f - mx[i]) * BIGPEN_;
    packed[i] = 0xFFFFFFFFFFFFFFFFull;
}

// Fused distance GEMM + argmin. One wave per (b, 32-row y-strip, x-slice).
// Two A strips resident in 128 VGPRs; B fragments streamed through an explicit
// 4-deep rotating buffer so 4 b64 loads stay in flight per wave.
__global__ __launch_bounds__(128) void k_argmin(
        const float* __restrict__ ey, const float* __restrict__ ex,
        const float* __restrict__ sx, unsigned long long* __restrict__ packed) {
    const int lane = threadIdx.x & 31;
    const int wid  = blockIdx.x * (blockDim.x >> 5) + (threadIdx.x >> 5);
    const int s  = wid % SLICES_;
    const int t  = wid / SLICES_;
    const int ty = t % YT32_;
    const int b  = t / YT32_;
    const int row = lane & 15, hi = lane >> 4;
    const int y0 = ty * 32;
    const int ycl0 = min(y0 + row, N_ - 1);
    const int ycl1 = min(y0 + 16 + row, N_ - 1);

    // A strips (2 x 16rows x 128k), resident in 128 VGPRs
    v2f a0[32], a1[32];
    const float* ar0 = ey + ((size_t)b * N_ + ycl0) * K_ + 2 * hi;
    const float* ar1 = ey + ((size_t)b * N_ + ycl1) * K_ + 2 * hi;
#pragma unroll
    for (int kc = 0; kc < 32; ++kc) {
        a0[kc] = *(const v2f*)(ar0 + kc * 4);
        a1[kc] = *(const v2f*)(ar1 + kc * 4);
    }

    float mv[16];
    int   mi[16];
#pragma unroll
    for (int r = 0; r < 16; ++r) { mv[r] = 3.0e38f; mi[r] = 0; }

    const int xt0 = s * 55;
    const int xt1 = min(xt0 + 55, XT_);
    for (int xt = xt0; xt < xt1; ++xt) {
        const int x   = xt * 16 + row;          // this lane's output column
        const int xcl = min(x, N_ - 1);
        const float sxv = (x < N_) ? sx[b * N_ + x] : 3.0e37f;
        const float* brow = ex + ((size_t)b * N_ + xcl) * K_ + 2 * hi;
        v8f c0 = {0.f, 0.f, 0.f, 0.f, 0.f, 0.f, 0.f, 0.f};
        v8f c1 = {0.f, 0.f, 0.f, 0.f, 0.f, 0.f, 0.f, 0.f};
        // 4-deep software pipeline on the streamed B fragments
        v2f bf[4];
#pragma unroll
        for (int p = 0; p < 4; ++p) bf[p] = *(const v2f*)(brow + p * 4);
#pragma unroll
        for (int kc = 0; kc < 32; ++kc) {
            v2f cur = bf[kc & 3];
            if (kc < 28) bf[kc & 3] = *(const v2f*)(brow + (kc + 4) * 4);
            c0 = wmma4(a0[kc], cur, c0);
            c1 = wmma4(a1[kc], cur, c1);
        }
#pragma unroll
        for (int r = 0; r < 8; ++r) {
            float s0 = sxv - 2.0f * c0[r];
            float s1 = sxv - 2.0f * c1[r];
            if (s0 < mv[r])     { mv[r]     = s0; mi[r]     = x; }
            if (s1 < mv[8 + r]) { mv[8 + r] = s1; mi[8 + r] = x; }
        }
    }
    // reduce across the 16 lanes of each half (cols), keep smallest idx on tie
#pragma unroll
    for (int m = 1; m <= 8; m <<= 1) {
#pragma unroll
        for (int r = 0; r < 16; ++r) {
            float ov = __shfl_xor(mv[r], m, 32);
            int   oi = __shfl_xor(mi[r], m, 32);
            if (ov < mv[r] || (ov == mv[r] && oi < mi[r])) { mv[r] = ov; mi[r] = oi; }
        }
    }
    if ((lane & 15) == 0) {
#pragma unroll
        for (int r = 0; r < 8; ++r) {
            int y = y0 + r + 8 * hi;
            if (y < N_) {
                unsigned long long pk =
                    ((unsigned long long)fkey(mv[r]) << 32) | (unsigned int)mi[r];
                atomicMin(&packed[b * N_ + y], pk);
            }
            int y2 = y0 + 16 + r + 8 * hi;
            if (y2 < N_) {
                unsigned long long pk =
                    ((unsigned long long)fkey(mv[8 + r]) << 32) | (unsigned int)mi[8 + r];
                atomicMin(&packed[b * N_ + y2], pk);
            }
        }
    }
}

__global__ void k_finalize(const unsigned long long* __restrict__ packed,
                           const float* __restrict__ my, int* __restrict__ p2p) {
    int i = blockIdx.x * blockDim.x + threadIdx.x;
    if (i >= B_ * N_) return;
    p2p[i] = (my[i] > 0.5f) ? (int)(packed[i] & 0xFFFFFFFFull) : 0;
}

// G[b,y,j] = ex[b, p2p[b,y], j]
__global__ void k_gather(const float* __restrict__ ex, const int* __restrict__ p2p,
                         float* __restrict__ G) {
    size_t i = (size_t)blockIdx.x * blockDim.x + threadIdx.x;
    if (i >= (size_t)B_ * N_ * K_) return;
    int j = (int)(i % K_);
    int y = (int)((i / K_) % N_);
    int b = (int)(i / ((size_t)N_ * K_));
    G[i] = ex[((size_t)b * N_ + p2p[b * N_ + y]) * K_ + j];
}

// Cyx_s[b] (128x128) = ety[b] (128 x 7000) @ G[b] (7000 x 128). One 16x16 tile/wave.
__global__ __launch_bounds__(128) void k_cyx(const float* __restrict__ ety,
                                             const float* __restrict__ G,
                                             float* __restrict__ cyxs) {
    const int lane = threadIdx.x & 31;
    const int wid  = blockIdx.x * (blockDim.x >> 5) + (threadIdx.x >> 5);  // 0..127
    const int b  = wid >> 6;
    const int t  = wid & 63;
    const int m0 = (t >> 3) * 16, j0 = (t & 7) * 16;
    const int row = lane & 15, hi = lane >> 4;
    const float* arow = ety + ((size_t)b * K_ + m0 + row) * (size_t)N_ + 2 * hi;
    v8f c = {0.f, 0.f, 0.f, 0.f, 0.f, 0.f, 0.f, 0.f};
    for (int kc = 0; kc < N_ / 4; ++kc) {
        v2f a = *(const v2f*)(arow + kc * 4);
        int kk = kc * 4 + 2 * hi;
        v2f bf;
        bf.x = G[((size_t)b * N_ + kk    ) * K_ + j0 + row];
        bf.y = G[((size_t)b * N_ + kk + 1) * K_ + j0 + row];
        c = wmma4(a, bf, c);
    }
#pragma unroll
    for (int r = 0; r < 8; ++r)
        cyxs[(size_t)b * K_ * K_ + (m0 + r + 8 * hi) * K_ + j0 + row] = c[r];
}

// Tt[b] (NT x K, transposed!) : Tt[x][m] = sum_j Cyx_s[m][j] * etx[j][x].
// Transposed layout makes k_big's B fragments single contiguous b64 loads.
__global__ __launch_bounds__(128) void k_tmat(const float* __restrict__ cyxs,
                                              const float* __restrict__ etx,
                                              float* __restrict__ Tt) {
    const int lane = threadIdx.x & 31;
    const int wid  = blockIdx.x * (blockDim.x >> 5) + (threadIdx.x >> 5);
    const int b  = wid / (8 * XT_);
    const int rr = wid % (8 * XT_);
    const int m0 = (rr / XT_) * 16;
    const int x0 = (rr % XT_) * 16;
    const int row = lane & 15, hi = lane >> 4;
    const int x   = x0 + row;
    const int xcl = min(x, N_ - 1);
    const float* arow = cyxs + ((size_t)b * K_ + m0 + row) * K_ + 2 * hi;
    v8f c = {0.f, 0.f, 0.f, 0.f, 0.f, 0.f, 0.f, 0.f};
#pragma unroll
    for (int kc = 0; kc < 32; ++kc) {
        v2f a = *(const v2f*)(arow + kc * 4);
        int kk = kc * 4 + 2 * hi;
        v2f bf;
        bf.x = etx[((size_t)b * K_ + kk    ) * (size_t)N_ + xcl];
        bf.y = etx[((size_t)b * K_ + kk + 1) * (size_t)N_ + xcl];
        c = wmma4(a, bf, c);
    }
#pragma unroll
    for (int r = 0; r < 8; ++r)
        Tt[(size_t)b * NT_ * K_ + (size_t)x * K_ + m0 + r + 8 * hi] = c[r];
}

// Pyx = Ey @ T, 32x32 per wave (2x2 WMMA tiles), A strips register-resident,
// B streamed via explicit double buffers (4 b64 loads in flight);
// Pxy written via LDS transpose so both outputs store coalesced.
__global__ __launch_bounds__(128) void k_big(const float* __restrict__ ey,
                                             const float* __restrict__ Tt,
                                             float* __restrict__ out) {
    __shared__ float lds[4][32][33];
    const size_t NN = (size_t)N_ * N_;
    const size_t O_PXY = 2 * (size_t)B_ * K_ * K_;
    const size_t O_PYX = O_PXY + (size_t)B_ * NN;

    const int lane = threadIdx.x & 31;
    const int wv   = threadIdx.x >> 5;
    const int wid  = blockIdx.x * 4 + wv;
    const bool active = wid < B_ * TL_ * TL_;
    const int w  = active ? wid : 0;
    const int b  = w / (TL_ * TL_);
    const int tt = w % (TL_ * TL_);
    const int Y0 = (tt / TL_) * 32;
    const int X0 = (tt % TL_) * 32;
    const int row = lane & 15, hi = lane >> 4;

    const float* abase = ey + (size_t)b * N_ * K_;
    const float* tbase = Tt + (size_t)b * NT_ * K_;
    const int ycl0 = min(Y0 + row, N_ - 1);
    const int ycl1 = min(Y0 + 16 + row, N_ - 1);

    // A strips resident in 128 VGPRs
    v2f a0f[32], a1f[32];
    const float* ar0 = abase + (size_t)ycl0 * K_ + 2 * hi;
    const float* ar1 = abase + (size_t)ycl1 * K_ + 2 * hi;
#pragma unroll
    for (int kc = 0; kc < 32; ++kc) {
        a0f[kc] = *(const v2f*)(ar0 + kc * 4);
        a1f[kc] = *(const v2f*)(ar1 + kc * 4);
    }

    v8f acc[2][2] = {};
    const float* b0r = tbase + (size_t)(X0 + row)      * K_ + 2 * hi;
    const float* b1r = tbase + (size_t)(X0 + 16 + row) * K_ + 2 * hi;
    // double-buffered B streams: 4 b64 loads in flight
    v2f b0buf[2], b1buf[2];
#pragma unroll
    for (int p = 0; p < 2; ++p) {
        b0buf[p] = *(const v2f*)(b0r + p * 4);
        b1buf[p] = *(const v2f*)(b1r + p * 4);
    }
#pragma unroll
    for (int kc = 0; kc < 32; ++kc) {
        v2f b0 = b0buf[kc & 1];
        v2f b1 = b1buf[kc & 1];
        if (kc < 30) {
            b0buf[kc & 1] = *(const v2f*)(b0r + (kc + 2) * 4);
            b1buf[kc & 1] = *(const v2f*)(b1r + (kc + 2) * 4);
        }
        acc[0][0] = wmma4(a0f[kc], b0, acc[0][0]);
        acc[0][1] = wmma4(a0f[kc], b1, acc[0][1]);
        acc[1][0] = wmma4(a1f[kc], b0, acc[1][0]);
        acc[1][1] = wmma4(a1f[kc], b1, acc[1][1]);
    }

    float* pyx = out + O_PYX + (size_t)b * NN;
    float* pxy = out + O_PXY + (size_t)b * NN;
#pragma unroll
    for (int i = 0; i < 2; ++i) {
#pragma unroll
        for (int j = 0; j < 2; ++j) {
#pragma unroll
            for (int r = 0; r < 8; ++r) {
                int ly = i * 16 + r + 8 * hi;
                int lx = j * 16 + row;
                int y = Y0 + ly, x = X0 + lx;
                if (active && y < N_ && x < N_)
                    pyx[(size_t)y * N_ + x] = acc[i][j][r];     // coalesced rows
                lds[wv][ly][lx] = acc[i][j][r];
            }
        }
    }
    __syncthreads();
#pragma unroll 4
    for (int rr = 0; rr < 32; ++rr) {
        int x = X0 + rr, y = Y0 + lane;
        if (active && x < N_ && y < N_)
            pxy[(size_t)x * N_ + y] = lds[wv][lane][rr];        // coalesced rows
    }
}

// Cxy and Cyx outputs are identity matrices.
__global__ void k_ident(float* __restrict__ out) {
    int i = blockIdx.x * blockDim.x + threadIdx.x;
    if (i >= 2 * B_ * K_ * K_) return;
    int w = i % (K_ * K_);
    out[i] = (w / K_ == w % K_) ? 1.0f : 0.0f;
}

extern "C" void kernel_launch(void* const* d_in, const int* in_sizes, int n_in,
                              void* d_out, int out_size, void* d_ws, size_t ws_size,
                              hipStream_t stream) {
    const float* ex  = (const float*)d_in[0];  // evecs_x       [B,N,K]
    const float* ey  = (const float*)d_in[1];  // evecs_y       [B,N,K]
    const float* etx = (const float*)d_in[2];  // evecs_trans_x [B,K,N]
    const float* ety = (const float*)d_in[3];  // evecs_trans_y [B,K,N]
    const float* mx  = (const float*)d_in[4];  // verts_mask_x  [B,N]
    const float* my  = (const float*)d_in[5];  // verts_mask_y  [B,N]
    float* out = (float*)d_out;

    char* ws = (char*)d_ws;
    size_t o = 0;
    auto take = [&](size_t bytes) {
        void* p = ws + o;
        o += (bytes + 255) & ~(size_t)255;
        return p;
    };
    float* sx                   = (float*)take((size_t)B_ * N_ * 4);
    unsigned long long* packed  = (unsigned long long*)take((size_t)B_ * N_ * 8);
    int* p2p                    = (int*)take((size_t)B_ * N_ * 4);
    float* G                    = (float*)take((size_t)B_ * N_ * K_ * 4);
    float* cyxs                 = (float*)take((size_t)B_ * K_ * K_ * 4);
    float* Tt                   = (float*)take((size_t)B_ * NT_ * K_ * 4);
    (void)ws_size; (void)in_sizes; (void)n_in; (void)out_size;

    k_prep<<<(B_ * N_ + 255) / 256, 256, 0, stream>>>(ex, mx, sx, packed);
    k_argmin<<<(B_ * YT32_ * SLICES_) / 4, 128, 0, stream>>>(ey, ex, sx, packed);
    k_finalize<<<(B_ * N_ + 255) / 256, 256, 0, stream>>>(packed, my, p2p);
    k_gather<<<(int)(((size_t)B_ * N_ * K_ + 255) / 256), 256, 0, stream>>>(ex, p2p, G);
    k_cyx<<<32, 128, 0, stream>>>(ety, G, cyxs);
    k_tmat<<<(B_ * 8 * XT_) / 4, 128, 0, stream>>>(cyxs, etx, Tt);
    k_big<<<(B_ * TL_ * TL_ + 3) / 4, 128, 0, stream>>>(ey, Tt, out);
    k_ident<<<(2 * B_ * K_ * K_ + 255) / 256, 256, 0, stream>>>(out);
}